// SetEmbed_56221121905193
// MI455X (gfx1250) — compile-verified
//
#include <hip/hip_runtime.h>

// ---------------------------------------------------------------------------
// SetEmbed (Deep Sets) on MI455X / gfx1250.
//
// Roofline: X = 2048*512*64 fp32 = 256 MB -> ~11 us @ 23.3 TB/s HBM.
// GEMM1 = 2.15 GFLOP fp32 via V_WMMA_F32_16X16X4_F32 -> stays memory-bound.
// W2 is hoisted past the linear masked sum-pool, eliminating the per-row
// second GEMM:  out = (sum_{s<n} relu(X@W1+b1)) @ W2 + n*b2) @ W3 + b3.
// ---------------------------------------------------------------------------

typedef __attribute__((ext_vector_type(2))) float v2f;
typedef __attribute__((ext_vector_type(8))) float v8f;

#define BB   2048
#define SS   512
#define NFT  64
#define KK1  16
#define KK2  12
#define NOUT 32

__global__ __launch_bounds__(256) void setembed_wmma_kernel(
    const float* __restrict__ X,      // [B,S,NFT]
    const int*   __restrict__ setSizes,
    const float* __restrict__ W1,     // [NFT,K1]
    const float* __restrict__ b1,     // [K1]
    const float* __restrict__ W2,     // [K1,K2]
    const float* __restrict__ b2,     // [K2]
    const float* __restrict__ W3,     // [K2,NOUT]
    const float* __restrict__ b3,     // [NOUT]
    float*       __restrict__ out)    // [B,NOUT]
{
    __shared__ float part[8][32];   // per-wave partial pooled sums
    __shared__ float pooled16[16];  // sum_{s<size} relu(X@W1+b1), per feature
    __shared__ float p12[KK2];      // pooled16 @ W2 + size*b2

    const int b    = blockIdx.x;
    const int tid  = threadIdx.x;
    const int wave = tid >> 5;
    const int lane = tid & 31;
    const int nm   = lane & 15;   // N index for B/C/D, M (row) index for A
    const int hi   = lane >> 4;   // lane half: selects K pair (A) / row+8 (C/D)

    const int setSize = setSizes[b];

    // --- Preload W1 into WMMA B-matrix layout registers (16 steps x 2 vgprs).
    // Step s, vgpr j: holds W1[4s + j + 2*hi][nm]  (B 4x16 striping).
    v2f w1b[16];
    #pragma unroll
    for (int s = 0; s < 16; ++s) {
        const int k0 = 4 * s + 2 * hi;
        w1b[s].x = W1[(k0 + 0) * KK1 + nm];
        w1b[s].y = W1[(k0 + 1) * KK1 + nm];
    }
    const float b1v = b1[nm];

    float acc = 0.0f;  // pooled partial: col=nm, rows {v + 8*hi} over all tiles

    // 4 iterations x 8 waves x 16 rows = S = 512 rows of this batch.
    #pragma unroll 1
    for (int it = 0; it < 4; ++it) {
        const int sBase = it * 128 + wave * 16;
        // A-matrix row pointer: lane `nm` owns row sBase+nm; upper lane half
        // reads K+2,K+3 (2*hi float offset). 8-byte aligned -> b64 loads.
        const float* xrow =
            X + ((size_t)b * SS + sBase + nm) * NFT + 2 * hi;

        v8f c = {};
        #pragma unroll
        for (int s = 0; s < 16; ++s) {
            v2f a = *(const v2f*)(xrow + 4 * s);   // global_load_b64, imm off
            // D = A(16x4) x B(4x16) + C, fp32 exact
            c = __builtin_amdgcn_wmma_f32_16x16x4_f32(
                    false, a, false, w1b[s], (short)0, c, false, false);
        }

        // c[v] = (X_tile @ W1)[row = v + 8*hi][col = nm]; bias+relu+mask+pool.
        #pragma unroll
        for (int v = 0; v < 8; ++v) {
            const float h    = fmaxf(c[v] + b1v, 0.0f);
            const int   srow = sBase + v + 8 * hi;
            acc += (srow < setSize) ? h : 0.0f;
        }
    }

    // --- Deterministic cross-wave reduction (no float atomics).
    part[wave][lane] = acc;
    __syncthreads();

    if (tid < 16) {
        float p = 0.0f;
        #pragma unroll
        for (int w = 0; w < 8; ++w)
            p += part[w][tid] + part[w][tid + 16];
        pooled16[tid] = p;
    }
    __syncthreads();

    // --- Hoisted W2:  p12 = pooled16 @ W2 + setSize * b2
    if (tid < KK2) {
        float p = (float)setSize * b2[tid];
        #pragma unroll
        for (int j = 0; j < KK1; ++j)
            p += pooled16[j] * W2[j * KK2 + tid];
        p12[tid] = p;
    }
    __syncthreads();

    // --- Final projection: out = p12 @ W3 + b3   (32 coalesced stores)
    if (tid < NOUT) {
        float o = b3[tid];
        #pragma unroll
        for (int k = 0; k < KK2; ++k)
            o += p12[k] * W3[k * NOUT + tid];
        out[(size_t)b * NOUT + tid] = o;
    }
}

extern "C" void kernel_launch(void* const* d_in, const int* in_sizes, int n_in,
                              void* d_out, int out_size, void* d_ws, size_t ws_size,
                              hipStream_t stream) {
    const float* X        = (const float*)d_in[0];
    const int*   setSizes = (const int*)  d_in[1];
    const float* W1       = (const float*)d_in[2];
    const float* b1       = (const float*)d_in[3];
    const float* W2       = (const float*)d_in[4];
    const float* b2       = (const float*)d_in[5];
    const float* W3       = (const float*)d_in[6];
    const float* b3       = (const float*)d_in[7];
    float*       out      = (float*)d_out;

    setembed_wmma_kernel<<<BB, 256, 0, stream>>>(
        X, setSizes, W1, b1, W2, b2, W3, b3, out);
}